// DisentangledHierarchicalEncoder_84009560309945
// MI455X (gfx1250) — compile-verified
//
#include <hip/hip_runtime.h>

// ---------------------------------------------------------------------------
// CDNA5 (gfx1250) implementation.
// Heavy work = catalog MLPs -> fp32 WMMA GEMMs (V_WMMA_F32_16X16X4_F32),
// full fp32 precision to match the fp32 reference.
// Global->LDS staging uses GLOBAL_LOAD_ASYNC_TO_LDS_B128 (ASYNCcnt) with
// double-buffered LDS tiles when the builtin is available.
// ---------------------------------------------------------------------------

typedef float v2f __attribute__((ext_vector_type(2)));
typedef float v8f __attribute__((ext_vector_type(8)));
typedef int v4i __attribute__((ext_vector_type(4)));

// pointer-to-AS1-v4i / pointer-to-AS3-v4i (attribute attaches to pointee)
typedef __attribute__((address_space(1))) v4i* as1_v4i_ptr;
typedef __attribute__((address_space(3))) v4i* as3_v4i_ptr;

#ifndef __has_builtin
#define __has_builtin(x) 0
#endif

#if __has_builtin(__builtin_amdgcn_global_load_async_to_lds_b128)
#define USE_ASYNC_LDS 1
#else
#define USE_ASYNC_LDS 0
#endif

#if USE_ASYNC_LDS
#if __has_builtin(__builtin_amdgcn_s_wait_asynccnt)
#define ASYNC_WAIT0() __builtin_amdgcn_s_wait_asynccnt(0)
#else
#define ASYNC_WAIT0() asm volatile("s_wait_asynccnt 0x0" ::: "memory")
#endif
#else
#define ASYNC_WAIT0() \
  do {                \
  } while (0)
#endif

// Copy 16B global -> LDS. Async (DMA, no VGPR data path) when available.
// Flat global address == AS1 address; low 32 bits of a flat LDS address are
// the LDS offset (ISA 10.2 aperture rules), so integer round-trips are exact.
__device__ __forceinline__ void copy16_g2l(const float* __restrict__ g,
                                           float* __restrict__ l) {
#if USE_ASYNC_LDS
  __builtin_amdgcn_global_load_async_to_lds_b128(
      (as1_v4i_ptr)(unsigned long long)g,
      (as3_v4i_ptr)(unsigned int)(unsigned long long)l, 0, 0);
#else
  float4 v = *reinterpret_cast<const float4*>(g);
  l[0] = v.x;
  l[1] = v.y;
  l[2] = v.z;
  l[3] = v.w;
#endif
}

__device__ __forceinline__ float wave_allreduce_sum(float v) {
#pragma unroll
  for (int off = 16; off > 0; off >>= 1) v += __shfl_xor(v, off, 32);
  return v;
}

// ------------------------- row inverse L2 norm -----------------------------
__global__ __launch_bounds__(256) void rownorm_kernel(
    const float* __restrict__ X, float* __restrict__ inv_out, int N, int D) {
  int row = blockIdx.x * 8 + (threadIdx.x >> 5);
  if (row >= N) return;
  int lane = threadIdx.x & 31;
  const float* x = X + (size_t)row * D;
  float ss = 0.f;
  for (int i = lane; i < D; i += 32) {
    float v = x[i];
    ss = fmaf(v, v, ss);
  }
  ss = wave_allreduce_sum(ss);
  if (lane == 0) inv_out[row] = 1.0f / fmaxf(sqrtf(ss), 1e-12f);
}

// ------------------------- fp32 WMMA GEMM ----------------------------------
// OUT[m,n] = act( inv[m] * (A[M,K] @ W[Nout,K]^T)[m,n] + bias[n] )
// (row L2-norm scale applied in the epilogue: diag(inv)*A*W^T == diag(inv)*(A*W^T))
// Block: 128 threads (4 wave32). Tile 64x64, K-step 32, double-buffered LDS
// filled by async global->LDS DMA; one barrier + one asynccnt wait per K-tile.
// Requires K % 32 == 0 and Nout % 64 == 0 (true for all layers here).
#define TM 64
#define TN 64
#define TK 32
#define LPAD 36  // padded LDS row stride (floats): 144B rows, 16B aligned

__global__ __launch_bounds__(128) void gemm_wmma_f32_kernel(
    const float* __restrict__ A, const float* __restrict__ W,
    const float* __restrict__ bias, const float* __restrict__ inv_norm,
    float* __restrict__ OUT, int M, int K, int Nout, int relu) {
  __shared__ alignas(16) float sA[2][TM * LPAD];
  __shared__ alignas(16) float sB[2][TN * LPAD];

  const int tid = threadIdx.x;
  const int lane = tid & 31;
  const int wv = tid >> 5;  // wave id 0..3
  const int m0 = blockIdx.x * TM;
  const int n0 = blockIdx.y * TN;

  const int ml = lane & 15;  // M (or N) index within a 16-wide fragment
  const int hh = lane >> 4;  // K-half selector for f32 WMMA fragments

  v8f acc[4];
#pragma unroll
  for (int t = 0; t < 4; ++t)
#pragma unroll
    for (int j = 0; j < 8; ++j) acc[t][j] = 0.0f;

  // Issue one 64x32 A tile + 64x32 B tile into LDS buffer `buf`.
  // Each thread moves 4+4 16-byte segments. OOB M-rows are skipped (their
  // stale LDS rows only feed accumulator rows that the epilogue masks off).
  auto issue_tiles = [&](int buf, int kb) {
#pragma unroll
    for (int i = 0; i < 4; ++i) {
      int f = i * 128 + tid;  // float4 segment id 0..511
      int r = f >> 3;         // tile row (8 segments per 32-float row)
      int c = (f & 7) * 4;    // col
      int gr = m0 + r;
      if (gr < M) copy16_g2l(A + (size_t)gr * K + kb + c, &sA[buf][r * LPAD + c]);
    }
#pragma unroll
    for (int i = 0; i < 4; ++i) {
      int f = i * 128 + tid;
      int r = f >> 3;
      int c = (f & 7) * 4;
      copy16_g2l(W + (size_t)(n0 + r) * K + kb + c, &sB[buf][r * LPAD + c]);
    }
  };

  const int nk = K / TK;
  issue_tiles(0, 0);

  int buf = 0;
  for (int it = 0; it < nk; ++it) {
    ASYNC_WAIT0();    // our DMA for `buf` has landed in LDS
    __syncthreads();  // everyone's landed; prev compute on buf^1 finished
    if (it + 1 < nk) issue_tiles(buf ^ 1, (it + 1) * TK);
#if !USE_ASYNC_LDS
    if (it + 1 < nk) __syncthreads();  // sync-path stores race compute reads
#endif

    // ---- 8 k-steps x 4 n-tiles = 32 WMMAs per wave per K-tile -----------
    const float* ar = &sA[buf][(wv * 16 + ml) * LPAD];
#pragma unroll
    for (int kk = 0; kk < TK; kk += 4) {
      v2f a;
      a.x = ar[kk + 2 * hh];
      a.y = ar[kk + 2 * hh + 1];
#pragma unroll
      for (int t = 0; t < 4; ++t) {
        const float* br = &sB[buf][(t * 16 + ml) * LPAD];
        v2f b;
        b.x = br[kk + 2 * hh];
        b.y = br[kk + 2 * hh + 1];
        acc[t] = __builtin_amdgcn_wmma_f32_16x16x4_f32(
            false, a, false, b, (short)0, acc[t], false, false);
      }
    }
    buf ^= 1;
  }

  // ---- epilogue: row scale + bias (+ReLU). D: m = j + 8*hh, n = ml -------
#pragma unroll
  for (int j = 0; j < 8; ++j) {
    int m = m0 + wv * 16 + j + 8 * hh;
    if (m >= M) continue;
    float sc = inv_norm ? inv_norm[m] : 1.0f;
#pragma unroll
    for (int t = 0; t < 4; ++t) {
      int n = n0 + t * 16 + ml;
      float val = fmaf(acc[t][j], sc, bias[n]);
      if (relu) val = fmaxf(val, 0.f);
      OUT[(size_t)m * Nout + n] = val;
    }
  }
}

// -------------- fused gather + L2norm + LayerNorm + attention --------------
// One wave32 per token; lane holds dims d=lane and d=lane+32.
__global__ __launch_bounds__(256) void fuse_attn_kernel(
    const int* __restrict__ seq, const float* __restrict__ ce,
    const float* __restrict__ te, const float* __restrict__ cfe,
    const float* __restrict__ ide, const float* __restrict__ wq,
    const float* __restrict__ wk, const float* __restrict__ wv,
    float* __restrict__ out, int n_tok, int num_item) {
  int tok = blockIdx.x * 8 + (threadIdx.x >> 5);
  if (tok >= n_tok) return;
  int lane = threadIdx.x & 31;

  int idx = seq[tok];
  if (idx == num_item) idx = 0;
  size_t base = (size_t)idx * 64;
  const float* srcs[4] = {ce + base, te + base, cfe + base, ide + base};

  float xn0[4], xn1[4];
#pragma unroll
  for (int m = 0; m < 4; ++m) {
    float a = srcs[m][lane];
    float b = srcs[m][lane + 32];
    // L2 normalize the modality vector
    float ss = wave_allreduce_sum(a * a + b * b);
    float inv = 1.0f / fmaxf(sqrtf(ss), 1e-12f);
    a *= inv;
    b *= inv;
    // LayerNorm (no affine)
    float mu = wave_allreduce_sum(a + b) * (1.0f / 64.0f);
    float ca = a - mu, cb = b - mu;
    float var = wave_allreduce_sum(ca * ca + cb * cb) * (1.0f / 64.0f);
    float r = rsqrtf(var + 1e-5f);
    xn0[m] = ca * r;
    xn1[m] = cb * r;
  }

  // q,k,v = xn @ W^T : broadcast xn[k] across lanes, each lane owns 2 rows.
  float q0[4] = {0, 0, 0, 0}, q1[4] = {0, 0, 0, 0};
  float k0[4] = {0, 0, 0, 0}, k1[4] = {0, 0, 0, 0};
  float v0[4] = {0, 0, 0, 0}, v1[4] = {0, 0, 0, 0};
  const float* wq0 = wq + (size_t)lane * 64;
  const float* wq1 = wq + (size_t)(lane + 32) * 64;
  const float* wk0 = wk + (size_t)lane * 64;
  const float* wk1 = wk + (size_t)(lane + 32) * 64;
  const float* wv0 = wv + (size_t)lane * 64;
  const float* wv1 = wv + (size_t)(lane + 32) * 64;

  for (int kx = 0; kx < 64; ++kx) {
    float aq0 = wq0[kx], aq1 = wq1[kx];
    float ak0 = wk0[kx], ak1 = wk1[kx];
    float av0 = wv0[kx], av1 = wv1[kx];
    int src = kx & 31;
    bool lo = kx < 32;
#pragma unroll
    for (int m = 0; m < 4; ++m) {
      float xv = __shfl(lo ? xn0[m] : xn1[m], src, 32);
      q0[m] = fmaf(xv, aq0, q0[m]);
      q1[m] = fmaf(xv, aq1, q1[m]);
      k0[m] = fmaf(xv, ak0, k0[m]);
      k1[m] = fmaf(xv, ak1, k1[m]);
      v0[m] = fmaf(xv, av0, v0[m]);
      v1[m] = fmaf(xv, av1, v1[m]);
    }
  }

  // scores s[m][n] = (q_m . k_n) / sqrt(64)
  float s[4][4];
#pragma unroll
  for (int m = 0; m < 4; ++m)
#pragma unroll
    for (int n = 0; n < 4; ++n)
      s[m][n] = wave_allreduce_sum(q0[m] * k0[n] + q1[m] * k1[n]) * 0.125f;

  // softmax over n, then mean over m -> per-column weight
  float wn[4] = {0, 0, 0, 0};
#pragma unroll
  for (int m = 0; m < 4; ++m) {
    float mx = fmaxf(fmaxf(s[m][0], s[m][1]), fmaxf(s[m][2], s[m][3]));
    float e[4];
    float sum = 0.f;
#pragma unroll
    for (int n = 0; n < 4; ++n) {
      e[n] = expf(s[m][n] - mx);
      sum += e[n];
    }
    float rs = 0.25f / sum;
#pragma unroll
    for (int n = 0; n < 4; ++n) wn[n] += e[n] * rs;
  }

  float o0 = 0.f, o1 = 0.f;
#pragma unroll
  for (int n = 0; n < 4; ++n) {
    o0 = fmaf(wn[n], v0[n], o0);
    o1 = fmaf(wn[n], v1[n], o1);
  }
  out[(size_t)tok * 64 + lane] = o0;
  out[(size_t)tok * 64 + lane + 32] = o1;
}

// ---------------------------------------------------------------------------
extern "C" void kernel_launch(void* const* d_in, const int* in_sizes, int n_in,
                              void* d_out, int out_size, void* d_ws,
                              size_t ws_size, hipStream_t stream) {
  const int* seq = (const int*)d_in[0];
  const float* content = (const float*)d_in[1];
  const float* text = (const float*)d_in[2];
  const float* cf = (const float*)d_in[3];
  const float* id_emb = (const float*)d_in[4];
  const float* c_w1 = (const float*)d_in[5];
  const float* c_b1 = (const float*)d_in[6];
  const float* c_w2 = (const float*)d_in[7];
  const float* c_b2 = (const float*)d_in[8];
  const float* c_w3 = (const float*)d_in[9];
  const float* c_b3 = (const float*)d_in[10];
  const float* t_w1 = (const float*)d_in[11];
  const float* t_b1 = (const float*)d_in[12];
  const float* t_w2 = (const float*)d_in[13];
  const float* t_b2 = (const float*)d_in[14];
  const float* t_w3 = (const float*)d_in[15];
  const float* t_b3 = (const float*)d_in[16];
  const float* cf_w = (const float*)d_in[17];
  const float* cf_b = (const float*)d_in[18];
  const float* wq = (const float*)d_in[19];
  const float* wk = (const float*)d_in[20];
  const float* wv = (const float*)d_in[21];

  const int d_c = 1024, d_t = 768;
  const int N = in_sizes[1] / d_c;  // 50000
  const int n_tok = in_sizes[0];    // 1024*50

  // workspace layout (floats); h1/h2 reused by content then text chain
  float* ws = (float*)d_ws;
  size_t off = 0;
  float* invC = ws + off; off += (size_t)N;
  float* invT = ws + off; off += (size_t)N;
  float* ceb = ws + off;  off += (size_t)N * 64;
  float* teb = ws + off;  off += (size_t)N * 64;
  float* cfe = ws + off;  off += (size_t)N * 64;
  float* h2 = ws + off;   off += (size_t)N * 256;
  float* h1 = ws + off;   off += (size_t)N * 1024;
  (void)ws_size; (void)n_in; (void)out_size;

  dim3 blk(128);
  auto grid = [](int M, int Nout) { return dim3((M + TM - 1) / TM, Nout / TN); };

  rownorm_kernel<<<(N + 7) / 8, 256, 0, stream>>>(content, invC, N, d_c);
  rownorm_kernel<<<(N + 7) / 8, 256, 0, stream>>>(text, invT, N, d_t);

  // content MLP: 1024 -> 1024 -> 256 -> 64
  gemm_wmma_f32_kernel<<<grid(N, 1024), blk, 0, stream>>>(
      content, c_w1, c_b1, invC, h1, N, 1024, 1024, 1);
  gemm_wmma_f32_kernel<<<grid(N, 256), blk, 0, stream>>>(
      h1, c_w2, c_b2, nullptr, h2, N, 1024, 256, 1);
  gemm_wmma_f32_kernel<<<grid(N, 64), blk, 0, stream>>>(
      h2, c_w3, c_b3, nullptr, ceb, N, 256, 64, 0);

  // text MLP: 768 -> 768 -> 256 -> 64 (reuses h1/h2)
  gemm_wmma_f32_kernel<<<grid(N, 768), blk, 0, stream>>>(
      text, t_w1, t_b1, invT, h1, N, 768, 768, 1);
  gemm_wmma_f32_kernel<<<grid(N, 256), blk, 0, stream>>>(
      h1, t_w2, t_b2, nullptr, h2, N, 768, 256, 1);
  gemm_wmma_f32_kernel<<<grid(N, 64), blk, 0, stream>>>(
      h2, t_w3, t_b3, nullptr, teb, N, 256, 64, 0);

  // cf linear: 64 -> 64
  gemm_wmma_f32_kernel<<<grid(N, 64), blk, 0, stream>>>(
      cf, cf_w, cf_b, nullptr, cfe, N, 64, 64, 0);

  // fused gather + norms + 4x4 attention + mean pool
  fuse_attn_kernel<<<(n_tok + 7) / 8, 256, 0, stream>>>(
      seq, ceb, teb, cfe, id_emb, wq, wk, wv, (float*)d_out, n_tok, N);
}